// Decoder1_25031069401694
// MI455X (gfx1250) — compile-verified
//
#include <hip/hip_runtime.h>
#include <hip/hip_bf16.h>

typedef float v8f __attribute__((ext_vector_type(8)));
typedef float v2f __attribute__((ext_vector_type(2)));

#define DIMD 128
#define NNODES 50000
#define NEDGES 800000

// ---------------------------------------------------------------------------
// z = H @ W   (H: [n,128] row-major, W: [128,128] row-major, z: [n,128])
// One wave -> one 16x16 tile of z. Block = 256 thr = 8 waves -> 16 rows x 128 cols.
// Uses V_WMMA_F32_16X16X4_F32, chaining K=128 through 32 WMMAs.
// ---------------------------------------------------------------------------
__global__ __launch_bounds__(256) void gemm_wmma_f32(
    const float* __restrict__ H, const float* __restrict__ W,
    float* __restrict__ Z) {
  const int wave = threadIdx.x >> 5;   // 0..7 -> column tile
  const int lane = threadIdx.x & 31;
  const int row0 = blockIdx.x * 16;
  const int col0 = wave * 16;
  const int m    = lane & 15;          // row within A tile / col within B tile
  const int kh   = (lane >> 4) * 2;    // 0 for lanes 0-15, 2 for lanes 16-31

  const float* __restrict__ arow = H + (size_t)(row0 + m) * DIMD;
  v8f acc = {};
#pragma unroll
  for (int k = 0; k < DIMD; k += 4) {
    v2f a, b;
    // A 16x4 layout: VGPR0 = K= kh, VGPR1 = K= kh+1  (ISA 7.12.2)
    a.x = arow[k + kh];
    a.y = arow[k + kh + 1];
    // B 4x16 layout: VGPR0 = row kh (this half of lanes), VGPR1 = row kh+1
    b.x = W[(size_t)(k + kh) * DIMD + col0 + m];
    b.y = W[(size_t)(k + kh + 1) * DIMD + col0 + m];
    acc = __builtin_amdgcn_wmma_f32_16x16x4_f32(
        /*neg_a=*/false, a, /*neg_b=*/false, b,
        /*c_mod=*/(short)0, acc, /*reuse_a=*/false, /*reuse_b=*/false);
  }
  // C/D layout: VGPR j -> M=j (lanes 0-15), M=j+8 (lanes 16-31)
  const int mbase = row0 + (lane >> 4) * 8;
#pragma unroll
  for (int j = 0; j < 8; ++j)
    Z[(size_t)(mbase + j) * DIMD + col0 + m] = acc[j];
}

// ---------------------------------------------------------------------------
// agg[dst[e]] += z[src[e]]   — one wave per edge, float4 per lane (512B/edge),
// f32 atomics land in L2 (both 25.6MB buffers are L2-resident on MI455X).
// ---------------------------------------------------------------------------
__global__ __launch_bounds__(256) void scatter_add_edges(
    const float* __restrict__ Z, const int* __restrict__ src,
    const int* __restrict__ dst, float* __restrict__ agg, int n_edges) {
  const int e    = blockIdx.x * 8 + (threadIdx.x >> 5);
  const int lane = threadIdx.x & 31;
  if (e >= n_edges) return;
  const int s = src[e];
  const int d = dst[e];
  const float4 v = ((const float4*)(Z + (size_t)s * DIMD))[lane];
  float* ap = agg + (size_t)d * DIMD + lane * 4;
  unsafeAtomicAdd(ap + 0, v.x);
  unsafeAtomicAdd(ap + 1, v.y);
  unsafeAtomicAdd(ap + 2, v.z);
  unsafeAtomicAdd(ap + 3, v.w);
}

// ---------------------------------------------------------------------------
// z = prelu(agg + b, a_conv) in place; accumulate per-column sum / sumsq.
// 128 threads = one thread per column; block handles 64 rows.
// ---------------------------------------------------------------------------
__global__ __launch_bounds__(128) void bias_prelu_stats(
    float* __restrict__ ZA, const float* __restrict__ bias,
    const float* __restrict__ a_conv_p, float* __restrict__ stats,
    int n_rows, int rows_per_block) {
  const int c  = threadIdx.x;  // column 0..127
  const float bb = bias[c];
  const float ac = *a_conv_p;
  const int r0 = blockIdx.x * rows_per_block;
  int r1 = r0 + rows_per_block;
  if (r1 > n_rows) r1 = n_rows;
  float s = 0.f, ss = 0.f;
  for (int r = r0; r < r1; ++r) {
    const size_t idx = (size_t)r * DIMD + c;
    float z = ZA[idx] + bb;
    z = (z >= 0.f) ? z : ac * z;
    ZA[idx] = z;
    s += z;
    ss += z * z;
  }
  unsafeAtomicAdd(&stats[c], s);
  unsafeAtomicAdd(&stats[DIMD + c], ss);
}

// ---------------------------------------------------------------------------
// Fold batch stats + gamma/beta into per-column scale/shift (in stats buffer).
// ---------------------------------------------------------------------------
__global__ __launch_bounds__(128) void bn_finalize(
    float* __restrict__ stats, const float* __restrict__ gamma,
    const float* __restrict__ beta, int n_rows) {
  const int c = threadIdx.x;
  const float inv_n = 1.0f / (float)n_rows;
  const float mu  = stats[c] * inv_n;
  const float var = stats[DIMD + c] * inv_n - mu * mu;
  const float scale = rsqrtf(var + 1e-5f) * gamma[c];
  const float shift = beta[c] - mu * scale;
  stats[c]        = scale;   // each thread owns its two slots: safe RMW
  stats[DIMD + c] = shift;
}

// ---------------------------------------------------------------------------
// h = prelu(z * scale[c] + shift[c], a_out)   — elementwise, grid-stride.
// ---------------------------------------------------------------------------
__global__ __launch_bounds__(256) void bn_apply_prelu(
    const float* __restrict__ Z, const float* __restrict__ stats,
    const float* __restrict__ a_out_p, float* __restrict__ Hout,
    size_t n_elems) {
  const float ao = *a_out_p;
  const size_t stride = (size_t)gridDim.x * blockDim.x;
  for (size_t i = (size_t)blockIdx.x * blockDim.x + threadIdx.x; i < n_elems;
       i += stride) {
    const int c = (int)(i & (DIMD - 1));
    float z = fmaf(Z[i], stats[c], stats[DIMD + c]);
    Hout[i] = (z >= 0.f) ? z : ao * z;
  }
}

extern "C" void kernel_launch(void* const* d_in, const int* in_sizes, int n_in,
                              void* d_out, int out_size, void* d_ws,
                              size_t ws_size, hipStream_t stream) {
  const float* heat   = (const float*)d_in[0];
  const int*   src    = (const int*)d_in[1];
  const int*   dst    = (const int*)d_in[2];
  const float* W      = (const float*)d_in[3];  // [2,128,128]
  const float* b      = (const float*)d_in[4];  // [2,128]
  const float* a_conv = (const float*)d_in[5];  // [2]
  const float* gamma  = (const float*)d_in[6];  // [2,128]
  const float* beta   = (const float*)d_in[7];  // [2,128]
  const float* a_out  = (const float*)d_in[8];  // [2]

  const size_t nb = (size_t)NNODES * DIMD * sizeof(float);
  float* bufA  = (float*)d_ws;
  float* bufB  = (float*)((char*)d_ws + nb);
  float* stats = (float*)((char*)d_ws + 2 * nb);

  const dim3 gemmGrid(NNODES / 16);       // 3125 tiles of 16 rows
  const dim3 gemmBlock(256);
  const int rows_per_block = 64;
  const int statBlocks = (NNODES + rows_per_block - 1) / rows_per_block;
  const int scatBlocks = (NEDGES + 7) / 8;
  const size_t nelem = (size_t)NNODES * DIMD;

  // ---------------- layer 0 ----------------
  hipMemsetAsync(bufB, 0, nb, stream);
  hipMemsetAsync(stats, 0, 2 * DIMD * sizeof(float), stream);
  gemm_wmma_f32<<<gemmGrid, gemmBlock, 0, stream>>>(heat, W, bufA);
  scatter_add_edges<<<scatBlocks, 256, 0, stream>>>(bufA, src, dst, bufB,
                                                    NEDGES);
  bias_prelu_stats<<<statBlocks, 128, 0, stream>>>(bufB, b, a_conv, stats,
                                                   NNODES, rows_per_block);
  bn_finalize<<<1, 128, 0, stream>>>(stats, gamma, beta, NNODES);
  bn_apply_prelu<<<4096, 256, 0, stream>>>(bufB, stats, a_out, bufA, nelem);

  // ---------------- layer 1 ----------------
  gemm_wmma_f32<<<gemmGrid, gemmBlock, 0, stream>>>(bufA, W + DIMD * DIMD,
                                                    bufB);
  hipMemsetAsync(bufA, 0, nb, stream);  // bufA (h1) dead after GEMM above
  hipMemsetAsync(stats, 0, 2 * DIMD * sizeof(float), stream);
  scatter_add_edges<<<scatBlocks, 256, 0, stream>>>(bufB, src, dst, bufA,
                                                    NEDGES);
  bias_prelu_stats<<<statBlocks, 128, 0, stream>>>(bufA, b + DIMD, a_conv + 1,
                                                   stats, NNODES,
                                                   rows_per_block);
  bn_finalize<<<1, 128, 0, stream>>>(stats, gamma + DIMD, beta + DIMD, NNODES);
  bn_apply_prelu<<<4096, 256, 0, stream>>>(bufA, stats, a_out + 1,
                                           (float*)d_out, nelem);
}